// BaseEnergyFormPhysics_59115929862624
// MI455X (gfx1250) — compile-verified
//
#include <hip/hip_runtime.h>
#include <hip/hip_bf16.h>
#include <stdint.h>

// ---------------------------------------------------------------------------
// Neo-Hookean FEM energy on gfx1250 (MI455X).
//  - grad_uq contraction via V_WMMA_F32_16X16X4_F32 (full f32 precision)
//  - WMMA D stored f-major in LDS (Gsh[e][f][m]): each v8f accumulator is
//    8 contiguous floats -> two direct ds_store_b128 per accumulator, no
//    register repacking moves.
//  - wave32 __shfl_xor butterfly for the in-block reduction (fixed order,
//    deterministic; replaces the LDS tree / 8 barriers).
//  - bandwidth-bound (~102 MB traffic): single pass, coalesced conns/JxW/state,
//    deterministic two-stage reduction (no float atomics).
// ---------------------------------------------------------------------------

typedef __attribute__((ext_vector_type(2))) float v2f;
typedef __attribute__((ext_vector_type(8))) float v8f;

#define N_ELEMS_C   400000
#define NNPE_C      8
#define NQ_C        8
#define ND_C        3
#define NDOF_C      3
#define EPW         16              // elements per wave (WMMA N dimension)
#define WPB         8               // waves per block (256 threads, wave32)
#define EPB         (EPW * WPB)     // 128 elements per block
#define NBLOCKS     (N_ELEMS_C / EPB)  // 3125 exactly

__global__ __launch_bounds__(256)
void fem_energy_kernel(const float* __restrict__ params,
                       const float* __restrict__ us,          // [N_NODES,3]
                       const float* __restrict__ shape_grads, // [Q,NNPE,ND]
                       const float* __restrict__ JxWs,        // [E,Q]
                       const float* __restrict__ state_old,   // [E,Q]
                       const long long* __restrict__ conns,   // [E,NNPE] int64
                       float* __restrict__ out_state,         // d_out + 1
                       float* __restrict__ block_sums)        // d_ws
{
    // Bsh[wave][f][n][e]       : gathered nodal displacements -> WMMA B matrices
    // Gsh[wave][e][f][m=3q+d]  : WMMA D results, f-major so each v8f accumulator
    //                            lands as 8 contiguous floats (2x ds_store_b128)
    __shared__ float Bsh[WPB][NDOF_C][NNPE_C][EPW];   // 12 KB
    __shared__ float Gsh[WPB][EPW][NDOF_C][24];       // 36 KB
    __shared__ float wsum[WPB];

    const int tid  = threadIdx.x;
    const int wave = tid >> 5;
    const int lane = tid & 31;
    const int elemBase = blockIdx.x * EPB + wave * EPW;

    const float mu  = params[0];
    const float lam = params[1];

    __builtin_prefetch(&JxWs[(size_t)elemBase * NQ_C], 0, 0);  // global_prefetch_b8

    // ---- Phase 1: gather. 128 (e_local,n) pairs per wave, coalesced conns ----
    #pragma unroll
    for (int i = 0; i < 4; ++i) {
        const int p = lane + 32 * i;            // 0..127
        const int e_local = p >> 3;
        const int n = p & 7;
        const long long c = conns[(long long)elemBase * NNPE_C + p];  // coalesced
        const float* up = us + (size_t)c * NDOF_C;                    // gather
        Bsh[wave][0][n][e_local] = up[0];
        Bsh[wave][1][n][e_local] = up[1];
        Bsh[wave][2][n][e_local] = up[2];
    }
    __syncthreads();

    // ---- Phase 2: WMMA  D[m=(3q+d), e] = sum_n SG^T[m,n] * us_f[n,e] --------
    // A (16x4 f32) ISA layout: lanes 0-15 -> M=lane, VGPR0=K0, VGPR1=K1;
    //                          lanes 16-31 -> M=lane-16, VGPR0=K2, VGPR1=K3.
    const int mrow  = lane & 15;
    const int khalf = (lane >> 4) << 1;   // 0 or 2
    const int col   = lane & 15;          // B/D column (element index in tile)
    const int upper = (lane >> 4);        // 0: D rows r, 1: D rows r+8

    v2f A[2][2];                          // [row-block][k-tile]
    #pragma unroll
    for (int rb = 0; rb < 2; ++rb) {
        #pragma unroll
        for (int t = 0; t < 2; ++t) {
            const int m = rb * 16 + mrow;          // (q,d) row, valid if < 24
            #pragma unroll
            for (int v = 0; v < 2; ++v) {
                const int k = t * 4 + khalf + v;   // node index n
                float val = 0.0f;
                if (m < 24) {
                    const int q = m / 3, d = m % 3;
                    val = shape_grads[(q * NNPE_C + k) * ND_C + d];
                }
                A[rb][t][v] = val;
            }
        }
    }

    // B (4x16 f32) layout assumed mirrored from A: VGPR0 holds K0 (lanes 0-15)
    // / K2 (lanes 16-31); VGPR1 holds K1 / K3.
    v2f B0[NDOF_C], B1[NDOF_C];
    #pragma unroll
    for (int f = 0; f < NDOF_C; ++f) {
        B0[f][0] = Bsh[wave][f][0 + khalf][col];
        B0[f][1] = Bsh[wave][f][1 + khalf][col];
        B1[f][0] = Bsh[wave][f][4 + khalf][col];
        B1[f][1] = Bsh[wave][f][5 + khalf][col];
    }

    // Per-lane LDS base for the D store: element slab = 3*24 floats (288 B,
    // 16B-aligned). Lanes 16-31 hold rows m+8 -> fold upper*32 B into the base.
    float* gbase = &Gsh[wave][col][0][upper * 8];

    #pragma unroll
    for (int rb = 0; rb < 2; ++rb) {
        v8f D[NDOF_C];
        #pragma unroll
        for (int f = 0; f < NDOF_C; ++f) {
            v8f acc = {};
            acc = __builtin_amdgcn_wmma_f32_16x16x4_f32(
                      false, A[rb][0], false, B0[f], (short)0, acc, false, false);
            acc = __builtin_amdgcn_wmma_f32_16x16x4_f32(
                      false, A[rb][1], false, B1[f], (short)0, acc, false, false);
            D[f] = acc;
        }
        // D (16x16 f32) layout: VGPR r: lanes 0-15 -> M=r, N=lane;
        //                       lanes 16-31 -> M=r+8, N=lane-16.
        // f-major store: D[f][r] -> gbase[f*24 + rb*16 + r]; contiguous in r,
        // 16B aligned -> two ds_store_b128 straight from the accumulator regs.
        if (rb == 0 || lane < 16) {       // rb=1 valid rows (16..23) live in lanes 0-15
            #pragma unroll
            for (int f = 0; f < NDOF_C; ++f)
                #pragma unroll
                for (int r = 0; r < 8; ++r)
                    gbase[f * 24 + rb * 16 + r] = D[f][r];
        }
    }
    __syncthreads();

    // ---- Phase 3: per-(e,q) Neo-Hookean energy + quadrature reduce ----------
    // grad_uq[f][d] = Gp[f*24 + 3q + d]
    const int e_local = lane & 15;
    const int elem    = elemBase + e_local;
    const int q0      = upper * 4;             // lanes split the 8 q-points
    const float* Gp   = &Gsh[wave][e_local][0][0];
    float acc_e = 0.0f;
    #pragma unroll
    for (int qi = 0; qi < 4; ++qi) {
        const int q = q0 + qi;
        const float* G0 = Gp + 3 * q;          // f = 0 row
        const float* G1 = G0 + 24;             // f = 1 row
        const float* G2 = G0 + 48;             // f = 2 row
        const float F00 = G0[0] + 1.0f, F01 = G0[1],        F02 = G0[2];
        const float F10 = G1[0],        F11 = G1[1] + 1.0f, F12 = G1[2];
        const float F20 = G2[0],        F21 = G2[1],        F22 = G2[2] + 1.0f;
        const float det = F00 * (F11 * F22 - F12 * F21)
                        - F01 * (F10 * F22 - F12 * F20)
                        + F02 * (F10 * F21 - F11 * F20);
        const float I1  = F00*F00 + F01*F01 + F02*F02
                        + F10*F10 + F11*F11 + F12*F12
                        + F20*F20 + F21*F21 + F22*F22;
        const float logJ = logf(det);
        const float fs = 0.5f * mu * (I1 - 3.0f - 2.0f * logJ)
                       + 0.5f * lam * logJ * logJ;
        acc_e += JxWs[(size_t)elem * NQ_C + q] * fs;
    }

    // ---- Phase 4: deterministic reduction: wave32 butterfly -> 8 partials ----
    #pragma unroll
    for (int m = 16; m > 0; m >>= 1)
        acc_e += __shfl_xor(acc_e, m, 32);
    if (lane == 0) wsum[wave] = acc_e;
    __syncthreads();
    if (tid == 0) {
        float s = 0.0f;
        #pragma unroll
        for (int w = 0; w < WPB; ++w) s += wsum[w];   // fixed order
        block_sums[blockIdx.x] = s;
    }

    // ---- Phase 5: state pass-through (1024 floats per block, coalesced) -----
    const size_t sbase = (size_t)blockIdx.x * EPB * NQ_C;
    #pragma unroll
    for (int i = 0; i < 4; ++i)
        out_state[sbase + tid + 256 * i] = state_old[sbase + tid + 256 * i];
}

// Fixed-order final reduction of the 3125 block partials -> pi (deterministic).
__global__ __launch_bounds__(256)
void final_reduce_kernel(const float* __restrict__ block_sums,
                         float* __restrict__ out)
{
    __shared__ float red[256];
    const int tid = threadIdx.x;
    float s = 0.0f;
    for (int i = tid; i < NBLOCKS; i += 256) s += block_sums[i];
    red[tid] = s;
    __syncthreads();
    #pragma unroll
    for (int st = 128; st > 0; st >>= 1) {
        if (tid < st) red[tid] += red[tid + st];
        __syncthreads();
    }
    if (tid == 0) out[0] = red[0];
}

extern "C" void kernel_launch(void* const* d_in, const int* in_sizes, int n_in,
                              void* d_out, int out_size, void* d_ws, size_t ws_size,
                              hipStream_t stream) {
    // Input order from setup_inputs():
    // 0 params, 1 coords(unused), 2 t(unused), 3 us, 4 shape_vals(unused),
    // 5 shape_grads, 6 JxWs, 7 dt(unused), 8 state_old, 9 conns(int64)
    const float*     params      = (const float*)d_in[0];
    const float*     us          = (const float*)d_in[3];
    const float*     shape_grads = (const float*)d_in[5];
    const float*     JxWs        = (const float*)d_in[6];
    const float*     state_old   = (const float*)d_in[8];
    const long long* conns       = (const long long*)d_in[9];

    float* out        = (float*)d_out;      // out[0] = pi, out[1..] = state_new
    float* block_sums = (float*)d_ws;       // NBLOCKS floats of scratch

    fem_energy_kernel<<<NBLOCKS, 256, 0, stream>>>(
        params, us, shape_grads, JxWs, state_old, conns, out + 1, block_sums);
    final_reduce_kernel<<<1, 256, 0, stream>>>(block_sums, out);
}